// DOTA_mix_36747740184748
// MI455X (gfx1250) — compile-verified
//
#include <hip/hip_runtime.h>

typedef __attribute__((ext_vector_type(2))) float v2f;
typedef __attribute__((ext_vector_type(8))) float v8f;

#define B_N    2048
#define K_N    1000
#define M_N    4
#define KM_N   4000
#define D_N    512
#define EPS_F    1e-3f
#define TINY_F   1e-10f
#define VARMIN_F 1e-8f

static __device__ __forceinline__ v8f wmma4(v2f a, v2f b, v8f c) {
    return __builtin_amdgcn_wmma_f32_16x16x4_f32(false, a, false, b, (short)0, c, false, false);
}

// CDNA5 async memory->LDS copies (ASYNCcnt-tracked), per CDNA5 ISA.
static __device__ __forceinline__ void async_ld_b128(const float* lds_dst, const float* gsrc) {
    unsigned lds = (unsigned)(uintptr_t)lds_dst;
    unsigned long long ga = (unsigned long long)(uintptr_t)gsrc;
    asm volatile("global_load_async_to_lds_b128 %0, %1, off" :: "v"(lds), "v"(ga) : "memory");
}
static __device__ __forceinline__ void async_ld_b32(const float* lds_dst, const float* gsrc) {
    unsigned lds = (unsigned)(uintptr_t)lds_dst;
    unsigned long long ga = (unsigned long long)(uintptr_t)gsrc;
    asm volatile("global_load_async_to_lds_b32 %0, %1, off" :: "v"(lds), "v"(ga) : "memory");
}
static __device__ __forceinline__ void wait_async0() {
    asm volatile("s_wait_asynccnt 0" ::: "memory");
}

// ---------------------------------------------------------------------------
// k1: per-(k,m) preparation: inv_var, mu*inv_var, sum mu^2*inv_var, log_det,
//     log(pi+TINY).  One workgroup per km row (4000 blocks x 256 threads).
// ---------------------------------------------------------------------------
__global__ __launch_bounds__(256) void k1_prep(
    const float* __restrict__ mu, const float* __restrict__ var,
    const float* __restrict__ pi,
    float* __restrict__ iv, float* __restrict__ miv,
    float* __restrict__ sq, float* __restrict__ ldet, float* __restrict__ lpi)
{
    const int km = blockIdx.x;
    const int t  = threadIdx.x;
    const size_t base = (size_t)km * D_N;
    float psq = 0.0f, pld = 0.0f;
    for (int d = t; d < D_N; d += 256) {
        float cv  = fmaxf(var[base + d] + EPS_F, VARMIN_F);
        float ivv = 1.0f / cv;
        float m   = mu[base + d];
        iv [base + d] = ivv;
        miv[base + d] = m * ivv;
        psq += m * m * ivv;
        pld += logf(cv);
    }
    __shared__ float s1[256], s2[256];
    s1[t] = psq; s2[t] = pld;
    __syncthreads();
    for (int o = 128; o > 0; o >>= 1) {
        if (t < o) { s1[t] += s1[t + o]; s2[t] += s2[t + o]; }
        __syncthreads();
    }
    if (t == 0) {
        sq[km]   = s1[0];
        ldet[km] = s2[0];
        lpi[km]  = logf(pi[km] + TINY_F);
    }
}

// ---------------------------------------------------------------------------
// k2: maha GEMM via v_wmma_f32_16x16x4_f32 + fused mode-softmax -> gamma.
// Workgroup tile: 256 b x 32 km, 8 waves; wave w owns a 32b x 32km register
// tile (8 v8f accumulators).  Tiles staged with async-to-LDS copies.
// ---------------------------------------------------------------------------
__global__ __launch_bounds__(256) void k2_resp(
    const float* __restrict__ x, const float* __restrict__ gclass,
    const float* __restrict__ iv, const float* __restrict__ miv,
    const float* __restrict__ sq, const float* __restrict__ ldet,
    const float* __restrict__ lpi,
    float* __restrict__ gamma, float* __restrict__ sum_gamma)
{
    __shared__ float lx [256 * 36];   // x tile   [256 b][32 d], padded stride 36
    __shared__ float liv[ 32 * 36];   // inv_var  [32 km][32 d]
    __shared__ float lmv[ 32 * 36];   // mu*iv    [32 km][32 d]
    __shared__ float ssg[32];         // per-block column sums of gamma

    const int t    = threadIdx.x;
    const int lane = t & 31;
    const int w    = t >> 5;          // wave id 0..7
    const int h    = lane >> 4;       // lane-half
    const int ln   = lane & 15;
    const int b0   = blockIdx.x * 256;
    const int km0  = blockIdx.y * 32;

    if (t < 32) ssg[t] = 0.0f;

    v8f acc1[2][2] = {};              // [b-subtile][km-subtile]: sum x^2*inv_var
    v8f acc2[2][2] = {};              // [b-subtile][km-subtile]: sum x*mu*inv_var

    for (int d0 = 0; d0 < D_N; d0 += 32) {
        __syncthreads();              // previous tile fully consumed
        // async stage x tile (256x32): 8 x 16B per thread, directly into LDS
#pragma unroll
        for (int i = 0; i < 8; ++i) {
            int idx = t + i * 256;
            int row = idx >> 3, c4 = (idx & 7) << 2;
            async_ld_b128(&lx[row * 36 + c4],
                          x + (size_t)(b0 + row) * D_N + d0 + c4);
        }
        // async stage inv_var / mu*inv_var tiles (32x32): 16B each per thread
        {
            int row = t >> 3, c4 = (t & 7) << 2;
            size_t g = (size_t)(km0 + row) * D_N + d0 + c4;
            async_ld_b128(&liv[row * 36 + c4], iv  + g);
            async_ld_b128(&lmv[row * 36 + c4], miv + g);
        }
        wait_async0();                // this wave's transfers landed in LDS
        __syncthreads();              // all waves' transfers visible
#pragma unroll
        for (int dd = 0; dd < 32; dd += 4) {
            const int dc = dd + 2 * h;
            // A fragments (16b x 4d) for the two b-subtiles
            v2f ax0 = *(const v2f*)&lx[(w * 32 +      ln) * 36 + dc];
            v2f ax1 = *(const v2f*)&lx[(w * 32 + 16 + ln) * 36 + dc];
            v2f as0 = ax0 * ax0;
            v2f as1 = ax1 * ax1;
            // B fragments (4d x 16km) for the two km-subtiles
            v2f bi0 = *(const v2f*)&liv[(     ln) * 36 + dc];
            v2f bi1 = *(const v2f*)&liv[(16 + ln) * 36 + dc];
            v2f bm0 = *(const v2f*)&lmv[(     ln) * 36 + dc];
            v2f bm1 = *(const v2f*)&lmv[(16 + ln) * 36 + dc];
            acc1[0][0] = wmma4(as0, bi0, acc1[0][0]);
            acc1[0][1] = wmma4(as0, bi1, acc1[0][1]);
            acc1[1][0] = wmma4(as1, bi0, acc1[1][0]);
            acc1[1][1] = wmma4(as1, bi1, acc1[1][1]);
            acc2[0][0] = wmma4(ax0, bm0, acc2[0][0]);
            acc2[0][1] = wmma4(ax0, bm1, acc2[0][1]);
            acc2[1][0] = wmma4(ax1, bm0, acc2[1][0]);
            acc2[1][1] = wmma4(ax1, bm1, acc2[1][1]);
        }
    }
    __syncthreads();

    // Epilogue: maha -> log-joint -> softmax over modes (4 adjacent km columns
    // == 4 adjacent lanes) -> gamma. C layout: VGPR r = batch row r + 8*h,
    // lane%16 = km column.
#pragma unroll
    for (int nt = 0; nt < 2; ++nt) {
        const int km = km0 + nt * 16 + ln;
        const float s_km  = sq[km];
        const float ld_km = ldet[km];
        const float lp_km = lpi[km];
        float colsum = 0.0f;
#pragma unroll
        for (int mt = 0; mt < 2; ++mt) {
#pragma unroll
            for (int r = 0; r < 8; ++r) {
                const int b = b0 + w * 32 + mt * 16 + r + 8 * h;
                float maha = acc1[mt][nt][r] - 2.0f * acc2[mt][nt][r] + s_km;
                float lj   = lp_km - 0.5f * (ld_km + maha);
                float mx = fmaxf(lj, __shfl_xor(lj, 1));
                mx       = fmaxf(mx, __shfl_xor(mx, 2));
                float e  = __expf(lj - mx);
                float se = e + __shfl_xor(e, 1);
                se      += __shfl_xor(se, 2);
                float gc = gclass[(size_t)b * K_N + (km >> 2)];
                float g  = gc * e / se;
                gamma[(size_t)b * KM_N + km] = g;
                colsum += g;
            }
        }
        colsum += __shfl_xor(colsum, 16);   // fold rows r+8 into lanes 0..15
        if (lane < 16) atomicAdd(&ssg[nt * 16 + ln], colsum);
    }
    __syncthreads();
    if (t < 32) atomicAdd(&sum_gamma[km0 + t], ssg[t]);
}

// ---------------------------------------------------------------------------
// k2b: c_new = c + sum_gamma ; pi_new = c_new / (row-sum over modes + TINY)
// ---------------------------------------------------------------------------
__global__ void k2b_cpi(const float* __restrict__ c,
                        const float* __restrict__ sum_gamma,
                        float* __restrict__ c_new, float* __restrict__ pi_new)
{
    int k = blockIdx.x * blockDim.x + threadIdx.x;
    if (k >= K_N) return;
    float cn[M_N]; float rs = 0.0f;
    for (int m = 0; m < M_N; ++m) { cn[m] = c[k * M_N + m] + sum_gamma[k * M_N + m]; rs += cn[m]; }
    float inv = 1.0f / (rs + TINY_F);
    for (int m = 0; m < M_N; ++m) { c_new[k * M_N + m] = cn[m]; pi_new[k * M_N + m] = cn[m] * inv; }
}

// ---------------------------------------------------------------------------
// k3: weighted_x / weighted_x_sq GEMM (gamma^T x) via WMMA, with the final
// mu_new / var_new elementwise update fused in the epilogue.
// Workgroup tile: 32 km x 256 d; wave w owns 32 km x 32 d (8 v8f accums).
// Full-B contraction; async-to-LDS staging (b32 scatter for the transpose).
// ---------------------------------------------------------------------------
__global__ __launch_bounds__(256) void k3_stats(
    const float* __restrict__ x, const float* __restrict__ gamma,
    const float* __restrict__ mu, const float* __restrict__ var,
    const float* __restrict__ c, const float* __restrict__ sum_gamma,
    float* __restrict__ mu_new, float* __restrict__ var_new)
{
    __shared__ float lg[32 * 36];    // gamma^T tile [32 km][32 b], stride 36
    __shared__ float lx[32 * 264];   // x tile       [32 b][256 d], stride 264

    const int t    = threadIdx.x;
    const int lane = t & 31;
    const int w    = t >> 5;         // d-slice 0..7
    const int h    = lane >> 4;
    const int ln   = lane & 15;
    const int km0  = blockIdx.x * 32;
    const int d0   = blockIdx.y * 256;

    v8f wx[2][2] = {};               // [km-subtile][d-subtile]: sum gamma*x
    v8f ws[2][2] = {};               // [km-subtile][d-subtile]: sum gamma*x^2

    for (int b0 = 0; b0 < B_N; b0 += 32) {
        __syncthreads();             // previous tile fully consumed
        // async stage gamma tile transposed to [km][b]: per-lane b32 scatter,
        // global reads coalesced over km
#pragma unroll
        for (int i = 0; i < 4; ++i) {
            int bi = (t >> 5) + i * 8;        // 0..31
            int ki = t & 31;                  // 0..31
            async_ld_b32(&lg[ki * 36 + bi],
                         gamma + (size_t)(b0 + bi) * KM_N + km0 + ki);
        }
        // async stage x tile [32 b][256 d]: 8 x 16B per thread
#pragma unroll
        for (int i = 0; i < 8; ++i) {
            int idx = t + i * 256;
            int row = idx >> 6, c4 = (idx & 63) << 2;
            async_ld_b128(&lx[row * 264 + c4],
                          x + (size_t)(b0 + row) * D_N + d0 + c4);
        }
        wait_async0();
        __syncthreads();
#pragma unroll
        for (int bb = 0; bb < 32; bb += 4) {
            // A fragments (16 km x 4 b) for the two km-subtiles
            v2f ag0 = *(const v2f*)&lg[(     ln) * 36 + bb + 2 * h];
            v2f ag1 = *(const v2f*)&lg[(16 + ln) * 36 + bb + 2 * h];
            // B fragments (4 b x 16 d) for the two d-subtiles
            const int dc = w * 32 + ln;
            const int ra = bb + 2 * h;
            v2f bx0 = { lx[ra * 264 + dc],      lx[(ra + 1) * 264 + dc]      };
            v2f bx1 = { lx[ra * 264 + dc + 16], lx[(ra + 1) * 264 + dc + 16] };
            v2f b20 = bx0 * bx0, b21 = bx1 * bx1;
            wx[0][0] = wmma4(ag0, bx0, wx[0][0]);
            wx[0][1] = wmma4(ag0, bx1, wx[0][1]);
            wx[1][0] = wmma4(ag1, bx0, wx[1][0]);
            wx[1][1] = wmma4(ag1, bx1, wx[1][1]);
            ws[0][0] = wmma4(ag0, b20, ws[0][0]);
            ws[0][1] = wmma4(ag0, b21, ws[0][1]);
            ws[1][0] = wmma4(ag1, b20, ws[1][0]);
            ws[1][1] = wmma4(ag1, b21, ws[1][1]);
        }
    }

    // Fused epilogue: each (km,d) output element computed exactly once here.
#pragma unroll
    for (int mt = 0; mt < 2; ++mt) {
#pragma unroll
        for (int nt = 0; nt < 2; ++nt) {
            const int d = d0 + w * 32 + nt * 16 + ln;
#pragma unroll
            for (int r = 0; r < 8; ++r) {
                const int km = km0 + mt * 16 + r + 8 * h;
                const float cv = c[km];
                const float sg = sum_gamma[km];
                const float denom = cv + sg + TINY_F;
                const size_t off = (size_t)km * D_N + d;
                const float m = mu[off], v = var[off];
                const float wxv = wx[mt][nt][r], wsv = ws[mt][nt][r];
                mu_new[off] = (cv * m + wxv) / denom;
                const float wsd = wsv - 2.0f * m * wxv + sg * m * m;
                var_new[off] = fmaxf((cv * v + wsd) / denom, VARMIN_F);
            }
        }
    }
}

// ---------------------------------------------------------------------------
extern "C" void kernel_launch(void* const* d_in, const int* in_sizes, int n_in,
                              void* d_out, int out_size, void* d_ws, size_t ws_size,
                              hipStream_t stream)
{
    (void)in_sizes; (void)n_in; (void)out_size; (void)ws_size;
    const float* x      = (const float*)d_in[0];
    const float* gclass = (const float*)d_in[1];
    const float* mu     = (const float*)d_in[2];
    const float* var    = (const float*)d_in[3];
    const float* pi     = (const float*)d_in[4];
    const float* c      = (const float*)d_in[5];

    float* out = (float*)d_out;
    float* ws  = (float*)d_ws;

    const size_t KMD = (size_t)KM_N * D_N;
    // workspace layout (floats): ~49.2 MB total
    float* iv    = ws;                 // KMD
    float* miv   = iv   + KMD;         // KMD
    float* sq    = miv  + KMD;         // KM
    float* ldet  = sq   + KM_N;        // KM
    float* lpi   = ldet + KM_N;        // KM
    float* sgm   = lpi  + KM_N;        // KM  (sum_gamma accumulator)
    float* gamma = sgm  + KM_N;        // B*KM (32 MB -- stays L2 resident)

    float* mu_new  = out;              // KMD
    float* var_new = out + KMD;        // KMD
    float* c_new   = out + 2 * KMD;    // KM
    float* pi_new  = c_new + KM_N;     // KM

    hipMemsetAsync(sgm, 0, KM_N * sizeof(float), stream);
    k1_prep<<<KM_N, 256, 0, stream>>>(mu, var, pi, iv, miv, sq, ldet, lpi);
    k2_resp<<<dim3(B_N / 256, KM_N / 32), 256, 0, stream>>>(
        x, gclass, iv, miv, sq, ldet, lpi, gamma, sgm);
    k2b_cpi<<<(K_N + 127) / 128, 128, 0, stream>>>(c, sgm, c_new, pi_new);
    k3_stats<<<dim3(KM_N / 32, D_N / 256), 256, 0, stream>>>(
        x, gamma, mu, var, c, sgm, mu_new, var_new);
}